// KLNearFar_37460704756321
// MI455X (gfx1250) — compile-verified
//
#include <hip/hip_runtime.h>
#include <hip/hip_bf16.h>
#include <math.h>

typedef __attribute__((ext_vector_type(2))) float v2f;
typedef __attribute__((ext_vector_type(8))) float v8f;

#define BLOCK 256          // 8 waves of 32
#define CHUNK 2048         // Y points staged per LDS round (32 KB)
#define ROWS_PER_WAVE 16
#define ROWS_PER_BLOCK 128 // 8 waves * 16 rows

// Chamfer min pass: for every point in X, min squared distance over all Y.
// d(i,j) = |x_i|^2 + (|y_j|^2 - 2 x_i.y_j), with the parenthesized part
// computed entirely inside one V_WMMA_F32_16X16X4_F32 per 16x16 tile:
//   A row  = (-2x0, -2x1, -2x2, 1)
//   B col  = ( y0,   y1,   y2,  |y|^2)
//   C      = |x|^2  (loop-invariant register octet)
// Y is staged in LDS pre-arranged in the native B VGPR layout so the inner
// loop is just ds_load_b64 + v_wmma + v_min per tile.
__global__ __launch_bounds__(BLOCK)
void chamfer_min_kernel(const float* __restrict__ Xall,
                        const float* __restrict__ Yall,
                        float* __restrict__ outmin,
                        int N, int M)
{
    // B-layout staging: per 16-point tile, 32 float2 entries.
    //   entry n      (n<16) = (y0[n], y1[n])   -> lanes 0..15  (K=0,1)
    //   entry 16+n          = (y2[n], |y|^2)   -> lanes 16..31 (K=2,3)
    __shared__ float2 sB[CHUNK * 2];

    const int b = blockIdx.y;
    const float* X = Xall + (size_t)b * N * 3;
    const float* Y = Yall + (size_t)b * M * 3;
    float* omin = outmin + (size_t)b * N;

    const int tid  = threadIdx.x;
    const int wave = tid >> 5;
    const int lane = tid & 31;
    const int n16  = lane & 15;
    const bool hi  = lane >= 16;

    const int xbase = blockIdx.x * ROWS_PER_BLOCK + wave * ROWS_PER_WAVE;

    // A matrix (16x4 f32): lane holds row m = n16, K = {2*hi, 1+2*hi}.
    const int xr = xbase + n16;
    const float x0  = X[xr * 3 + 0];
    const float x1  = X[xr * 3 + 1];
    const float x2c = X[xr * 3 + 2];
    v2f a;
    a.x = hi ? (-2.0f * x2c) : (-2.0f * x0);   // K=2 : K=0
    a.y = hi ? 1.0f          : (-2.0f * x1);   // K=3 : K=1

    // Loop-invariant C: |x|^2 for rows M = v + 8*hi (native C/D layout).
    v8f cmat;
#pragma unroll
    for (int v = 0; v < 8; ++v) {
        const int r = xbase + (hi ? 8 : 0) + v;
        const float c0 = X[r * 3 + 0];
        const float c1 = X[r * 3 + 1];
        const float c2 = X[r * 3 + 2];
        cmat[v] = c0 * c0 + c1 * c1 + c2 * c2;
    }

    float rmin[8];
#pragma unroll
    for (int v = 0; v < 8; ++v) rmin[v] = 3.0e38f;

    for (int cbase = 0; cbase < M; cbase += CHUNK) {
        __syncthreads();
        // Cooperative stage into B-layout LDS.
#pragma unroll
        for (int t = 0; t < CHUNK / BLOCK; ++t) {
            const int p    = tid + t * BLOCK;
            const int gp   = cbase + p;
            const int tile = p >> 4;
            const int n    = p & 15;
            const float y0 = Y[gp * 3 + 0];
            const float y1 = Y[gp * 3 + 1];
            const float y2 = Y[gp * 3 + 2];
            sB[tile * 32 + n]      = make_float2(y0, y1);
            sB[tile * 32 + 16 + n] = make_float2(y2, y0 * y0 + y1 * y1 + y2 * y2);
        }
        __syncthreads();

#pragma unroll 4
        for (int j = 0; j < CHUNK / 16; ++j) {
            const float2 bw = sB[j * 32 + lane];   // single ds_load_b64
            v2f bv;
            bv.x = bw.x;
            bv.y = bw.y;
            const v8f d = __builtin_amdgcn_wmma_f32_16x16x4_f32(
                false, a, false, bv, (short)0, cmat, false, false);
#pragma unroll
            for (int v = 0; v < 8; ++v) rmin[v] = fminf(rmin[v], d[v]);
        }
    }

    // Butterfly min across the 16 lanes of each half (tile columns).
#pragma unroll
    for (int v = 0; v < 8; ++v) {
        float r = rmin[v];
        r = fminf(r, __shfl_xor(r, 1, 32));
        r = fminf(r, __shfl_xor(r, 2, 32));
        r = fminf(r, __shfl_xor(r, 4, 32));
        r = fminf(r, __shfl_xor(r, 8, 32));
        rmin[v] = r;
    }
    if (lane == 0) {
#pragma unroll
        for (int v = 0; v < 8; ++v) omin[xbase + v] = rmin[v];
    } else if (lane == 16) {
#pragma unroll
        for (int v = 0; v < 8; ++v) omin[xbase + 8 + v] = rmin[v];
    }
}

// Fold both min arrays + the KL term into the scalar loss.
__global__ __launch_bounds__(1024)
void finalize_kernel(const float* __restrict__ minx,
                     const float* __restrict__ miny,
                     const float* __restrict__ mean,
                     const float* __restrict__ logvar,
                     float* __restrict__ out,
                     int nmin, int nkl, float inv_min, float inv_b)
{
    __shared__ float red[1024];
    const int tid = threadIdx.x;

    float s = 0.0f;
    for (int i = tid; i < nmin; i += 1024)
        s += (minx[i] + miny[i]) * inv_min;

    float skl = 0.0f;
    for (int i = tid; i < nkl; i += 1024) {
        const float m = mean[i];
        const float lv = logvar[i];
        skl += 0.5f * (m * m + expf(lv) - 1.0f - lv);
    }
    s += skl * inv_b;

    red[tid] = s;
    __syncthreads();
    for (int off = 512; off > 0; off >>= 1) {
        if (tid < off) red[tid] += red[tid + off];
        __syncthreads();
    }
    if (tid == 0) out[0] = red[0];
}

extern "C" void kernel_launch(void* const* d_in, const int* in_sizes, int n_in,
                              void* d_out, int out_size, void* d_ws, size_t ws_size,
                              hipStream_t stream) {
    const float* gt     = (const float*)d_in[0];   // [4,8192,3]
    const float* rc     = (const float*)d_in[1];   // [4,8192,3]
    const float* mean   = (const float*)d_in[2];   // [4,256,64]
    const float* logvar = (const float*)d_in[3];   // [4,256,64]

    const int B = 4;
    const int N = in_sizes[0] / (B * 3);   // 8192
    const int M = in_sizes[1] / (B * 3);   // 8192
    const int nkl = in_sizes[2];           // 65536

    float* minx = (float*)d_ws;            // [B*N]
    float* miny = minx + (size_t)B * N;    // [B*M]

    chamfer_min_kernel<<<dim3(N / ROWS_PER_BLOCK, B), BLOCK, 0, stream>>>(
        gt, rc, minx, N, M);
    chamfer_min_kernel<<<dim3(M / ROWS_PER_BLOCK, B), BLOCK, 0, stream>>>(
        rc, gt, miny, M, N);

    finalize_kernel<<<1, 1024, 0, stream>>>(
        minx, miny, mean, logvar, (float*)d_out,
        B * N, nkl, 1.0f / (float)(B * N), 1.0f / (float)B);
}